// BestKRouter_73753178407348
// MI455X (gfx1250) — compile-verified
//
#include <hip/hip_runtime.h>
#include <math.h>

typedef __attribute__((ext_vector_type(2))) float v2f;
typedef __attribute__((ext_vector_type(8))) float v8f;

#define BB 4
#define SS 4096
#define DD 2048
#define EE 64
#define NTOK (BB * SS)   // 16384 tokens
#define KSPLIT 2         // waves per block, each reduces DD/KSPLIT

// Two waves (64 threads) per 16-token tile; each wave reduces half of D with
// V_WMMA_F32_16X16X4_F32 into a partial 16x64 logit tile in LDS, then the
// partials are summed, top-2 + 2-way softmax computed, and scattered out.
__global__ __launch_bounds__(32 * KSPLIT) void BestKRouter_wmma_f32(
    const float* __restrict__ x,    // [NTOK, DD]
    const float* __restrict__ W,    // [EE, DD]
    const float* __restrict__ bias, // [EE]
    float* __restrict__ out,        // [NTOK, EE]
    int* __restrict__ idx_out)      // [NTOK, 2]
{
    __shared__ float logits[KSPLIT][16 * EE];
    __shared__ float p1s[16], p2s[16];
    __shared__ int   i1s[16], i2s[16];

    const int tid  = threadIdx.x;      // 0..63
    const int wave = tid >> 5;         // 0..1 : which D-half
    const int lane = tid & 31;         // 0..31
    const int col  = lane & 15;        // M (A) / N (B/C) index
    const int half = lane >> 4;        // K pair select: K = 2*half + vgpr
    const int tile = blockIdx.x;       // 0..1023
    const int token0 = tile * 16;

    // Base pointers already offset into this wave's D-half.
    const int kbase0 = wave * (DD / KSPLIT);
    const float* __restrict__ xrow = x + (size_t)(token0 + col) * DD + kbase0;
    const float* __restrict__ w0 = W + (size_t)(col +  0) * DD + kbase0;
    const float* __restrict__ w1 = W + (size_t)(col + 16) * DD + kbase0;
    const float* __restrict__ w2 = W + (size_t)(col + 32) * DD + kbase0;
    const float* __restrict__ w3 = W + (size_t)(col + 48) * DD + kbase0;

    v8f acc0 = {}, acc1 = {}, acc2 = {}, acc3 = {};

    // K-chunk loop over this wave's half: lane supplies K = kc + 2*half + {0,1}
    int k = 2 * half;
    #pragma unroll 4
    for (int kc = 0; kc < DD / KSPLIT; kc += 4, k += 4) {
        v2f a  = *(const v2f*)(xrow + k);
        v2f b0 = *(const v2f*)(w0 + k);
        v2f b1 = *(const v2f*)(w1 + k);
        v2f b2 = *(const v2f*)(w2 + k);
        v2f b3 = *(const v2f*)(w3 + k);
        acc0 = __builtin_amdgcn_wmma_f32_16x16x4_f32(false, a, false, b0, (short)0, acc0, false, false);
        acc1 = __builtin_amdgcn_wmma_f32_16x16x4_f32(false, a, false, b1, (short)0, acc1, false, false);
        acc2 = __builtin_amdgcn_wmma_f32_16x16x4_f32(false, a, false, b2, (short)0, acc2, false, false);
        acc3 = __builtin_amdgcn_wmma_f32_16x16x4_f32(false, a, false, b3, (short)0, acc3, false, false);
    }

    // C/D layout: VGPR r holds token m = r + 8*half, lane column col = expert%16.
    #pragma unroll
    for (int r = 0; r < 8; ++r) {
        const int m = r + 8 * half;
        logits[wave][m * EE + col +  0] = acc0[r];
        logits[wave][m * EE + col + 16] = acc1[r];
        logits[wave][m * EE + col + 32] = acc2[r];
        logits[wave][m * EE + col + 48] = acc3[r];
    }
    __syncthreads();

    // Threads 0..15: per-token top-2 over the summed partials (strict > keeps
    // lowest index on ties, matching jax.lax.top_k) and exact 2-way softmax.
    if (tid < 16) {
        const float* l0 = &logits[0][tid * EE];
        const float* l1 = &logits[1][tid * EE];
        float v1 = -INFINITY; int i1 = 0;
        for (int e = 0; e < EE; ++e) {
            float v = l0[e] + l1[e] + bias[e];
            if (v > v1) { v1 = v; i1 = e; }
        }
        float v2 = -INFINITY; int i2 = 0;
        for (int e = 0; e < EE; ++e) {
            if (e == i1) continue;
            float v = l0[e] + l1[e] + bias[e];
            if (v > v2) { v2 = v; i2 = e; }
        }
        // softmax over {v1, v2} with max subtraction; other entries exp(-inf)=0
        const float t  = __expf(v2 - v1);          // <= 1, no overflow
        const float p1 = 1.0f / (1.0f + t);
        const float p2 = t * p1;
        p1s[tid] = p1; p2s[tid] = p2;
        i1s[tid] = i1; i2s[tid] = i2;

        const int token = token0 + tid;
        idx_out[2 * token]     = i1;
        idx_out[2 * token + 1] = i2;
    }
    __syncthreads();

    // Coalesced writeback: 64 threads cover one full 64-float row per step.
    {
        const int e = tid;  // expert column for this thread
        #pragma unroll 4
        for (int m = 0; m < 16; ++m) {
            float v = (e == i1s[m]) ? p1s[m] : ((e == i2s[m]) ? p2s[m] : 0.0f);
            out[(size_t)(token0 + m) * EE + e] = v;
        }
    }
}

extern "C" void kernel_launch(void* const* d_in, const int* in_sizes, int n_in,
                              void* d_out, int out_size, void* d_ws, size_t ws_size,
                              hipStream_t stream) {
    const float* x    = (const float*)d_in[0];
    const float* W    = (const float*)d_in[1];
    const float* bias = (const float*)d_in[2];
    float* out = (float*)d_out;
    // Output tuple: router_out [4,4096,64] f32 followed by idxs [4,4096,2] int32.
    int* idx_out = (int*)(out + (size_t)NTOK * EE);

    BestKRouter_wmma_f32<<<NTOK / 16, 32 * KSPLIT, 0, stream>>>(x, W, bias, out, idx_out);
}